// ResidualBottleneck_19052474925461
// MI455X (gfx1250) — compile-verified
//
#include <hip/hip_runtime.h>

// ---------------------------------------------------------------------------
// ResidualBottleneck for MI455X (gfx1250, wave32, WMMA f32 16x16x4)
// B=8 T=4096 H=1024 D=256 Q=4 K=1024  -> M = B*T = 32768 rows
// ---------------------------------------------------------------------------

typedef __attribute__((ext_vector_type(2))) float v2f;
typedef __attribute__((ext_vector_type(8))) float v8f;

#define M_ROWS   32768
#define H_DIM    1024
#define D_DIM    256
#define Q_NUM    4
#define K_CODES  1024
#define ROW_TILES (M_ROWS / 16)   // 2048

__device__ __forceinline__ v8f wmma_f32(v2f a, v2f b, v8f c) {
  // D = A(16x4) * B(4x16) + C(16x16), fp32, wave32
  return __builtin_amdgcn_wmma_f32_16x16x4_f32(
      /*neg_a=*/false, a, /*neg_b=*/false, b,
      /*c_mod=*/(short)0, c, /*reuse_a=*/false, /*reuse_b=*/false);
}

__device__ __forceinline__ v2f ld2(const float* p) { return *(const v2f*)p; }

// ---------------------------------------------------------------------------
// per-code squared norms: cn[q*K + n] = sum_d cb[q][n][d]^2
// ---------------------------------------------------------------------------
__global__ __launch_bounds__(256) void cnorm_kernel(const float* __restrict__ cb,
                                                    float* __restrict__ cn) {
  const int i = blockIdx.x * 256 + threadIdx.x;      // [0, Q*K) = 4096
  const float4* r4 = (const float4*)(cb + (size_t)i * D_DIM);
  float s = 0.f;
  #pragma unroll 8
  for (int j = 0; j < D_DIM / 4; ++j) {
    float4 v = r4[j];
    s += v.x * v.x + v.y * v.y + v.z * v.z + v.w * v.w;
  }
  cn[i] = s;
}

// ---------------------------------------------------------------------------
// GEMM1: h[m][d] = sum_k x[m][k] * Win[d][k] + bin[d]
// One wave per 16-row tile; 4 N-tiles per A-fragment load.
// ---------------------------------------------------------------------------
__global__ __launch_bounds__(128) void gemm1_kernel(const float* __restrict__ x,
                                                    const float* __restrict__ Win,
                                                    const float* __restrict__ bin,
                                                    float* __restrict__ h) {
  const int lane  = threadIdx.x & 31;
  const int wave  = threadIdx.x >> 5;
  const int tile  = blockIdx.x * 4 + wave;
  const int row0  = tile << 4;
  const int rlane = lane & 15;
  const int khalf = (lane >> 4) << 1;           // 0 or 2: K split across halves
  const float* arow = x + (size_t)(row0 + rlane) * H_DIM + khalf;

  for (int ng = 0; ng < 4; ++ng) {              // 4 groups x 4 N-tiles = 256 cols
    const int n0 = ng << 6;
    const float* bp[4];
    #pragma unroll
    for (int j = 0; j < 4; ++j) {
      bp[j] = Win + (size_t)(n0 + (j << 4) + rlane) * H_DIM + khalf;
      __builtin_prefetch(bp[j], 0, 0);
    }
    v8f acc[4] = {{0,0,0,0,0,0,0,0},{0,0,0,0,0,0,0,0},
                  {0,0,0,0,0,0,0,0},{0,0,0,0,0,0,0,0}};
    #pragma unroll 4
    for (int k0 = 0; k0 < H_DIM; k0 += 4) {
      v2f a = ld2(arow + k0);
      #pragma unroll
      for (int j = 0; j < 4; ++j)
        acc[j] = wmma_f32(a, ld2(bp[j] + k0), acc[j]);
    }
    const int rbase = row0 + ((lane >> 4) << 3);
    #pragma unroll
    for (int j = 0; j < 4; ++j) {
      const int col  = n0 + (j << 4) + rlane;
      const float bv = bin[col];
      #pragma unroll
      for (int p = 0; p < 8; ++p)
        h[(size_t)(rbase + p) * D_DIM + col] = acc[j][p] + bv;
    }
  }
}

// ---------------------------------------------------------------------------
// Fused residual-VQ (4 rounds) + GEMM2.
// 2 waves/block, each wave owns one 16-row tile staged in LDS (16 KB/wave).
// Per round: distance GEMM via WMMA, cross-lane argmin, residual update,
// commit partial, q1/q2 emission (rounds 0/1). Then out = (h - res)*Wout^T+b.
// ---------------------------------------------------------------------------
__global__ __launch_bounds__(64) void vq_fused_kernel(
    const float* __restrict__ h, const float* __restrict__ cb,
    const float* __restrict__ cnorms, const float* __restrict__ Wout,
    const float* __restrict__ bout, float* __restrict__ out,
    float* __restrict__ q1, float* __restrict__ q2,
    float* __restrict__ commitbuf) {
  __shared__ float res[2][16 * D_DIM];   // residual tiles, one per wave
  __shared__ int   sidx[2][16];          // winning code index per row

  const int lane  = threadIdx.x & 31;
  const int wave  = threadIdx.x >> 5;
  const int tile  = blockIdx.x * 2 + wave;
  const int row0  = tile << 4;
  float* R = res[wave];

  { // stage h tile into LDS (residual starts as h)
    float4* R4 = (float4*)R;
    const float4* H4 = (const float4*)(h + (size_t)row0 * D_DIM);
    for (int i = lane; i < (16 * D_DIM) / 4; i += 32) R4[i] = H4[i];
  }
  __syncthreads();

  const int rlane = lane & 15;
  const int khalf = (lane >> 4) << 1;
  const float* aptr = R + rlane * D_DIM + khalf;

  for (int qi = 0; qi < Q_NUM; ++qi) {
    const float* cbq = cb + (size_t)qi * K_CODES * D_DIM;
    const float* cnq = cnorms + qi * K_CODES;

    float minv[8]; int mini[8];
    #pragma unroll
    for (int p = 0; p < 8; ++p) { minv[p] = 3.4e38f; mini[p] = 0; }

    for (int ng = 0; ng < K_CODES / 64; ++ng) {   // 16 groups of 4 N-tiles
      const int n0 = ng << 6;
      const float* bp[4];
      #pragma unroll
      for (int j = 0; j < 4; ++j) {
        bp[j] = cbq + (size_t)(n0 + (j << 4) + rlane) * D_DIM + khalf;
        __builtin_prefetch(bp[j], 0, 0);
      }
      v8f acc[4] = {{0,0,0,0,0,0,0,0},{0,0,0,0,0,0,0,0},
                    {0,0,0,0,0,0,0,0},{0,0,0,0,0,0,0,0}};
      #pragma unroll 4
      for (int k0 = 0; k0 < D_DIM; k0 += 4) {
        v2f a = ld2(aptr + k0);                   // A from LDS residual tile
        #pragma unroll
        for (int j = 0; j < 4; ++j)
          acc[j] = wmma_f32(a, ld2(bp[j] + k0), acc[j]);
      }
      #pragma unroll
      for (int j = 0; j < 4; ++j) {
        const int col = n0 + (j << 4) + rlane;
        const float cn = cnq[col];
        #pragma unroll
        for (int p = 0; p < 8; ++p) {
          const float dist = cn - 2.f * acc[j][p];  // ||c||^2 - 2 x.c
          if (dist < minv[p]) { minv[p] = dist; mini[p] = col; }
        }
      }
    }

    // argmin across the 16 lanes of each half (xor<16 stays within half)
    #pragma unroll
    for (int p = 0; p < 8; ++p) {
      #pragma unroll
      for (int off = 1; off < 16; off <<= 1) {
        const float ov = __shfl_xor(minv[p], off, 32);
        const int   oi = __shfl_xor(mini[p], off, 32);
        if (ov < minv[p] || (ov == minv[p] && oi < mini[p])) {
          minv[p] = ov; mini[p] = oi;
        }
      }
    }
    if (rlane == 0) {
      const int rb = (lane >> 4) << 3;
      #pragma unroll
      for (int p = 0; p < 8; ++p) sidx[wave][rb + p] = mini[p];
    }
    __syncthreads();

    // residual update + commit (= sum res_new^2) + q1/q2 emission
    const int c0 = lane << 3;                     // 8 cols per lane
    float csum = 0.f;
    for (int r = 0; r < 16; ++r) {
      const int idx = sidx[wave][r];
      const float* crow = cbq + (size_t)idx * D_DIM + c0;
      float* rrow = R + r * D_DIM + c0;
      float* qdst = (qi == 0) ? (q1 + (size_t)(row0 + r) * D_DIM + c0)
                  : (qi == 1) ? (q2 + (size_t)(row0 + r) * D_DIM + c0)
                              : (float*)0;
      #pragma unroll
      for (int c = 0; c < 8; ++c) {
        const float cv = crow[c];
        const float rv = rrow[c] - cv;
        rrow[c] = rv;
        csum += rv * rv;
        if (qdst) qdst[c] = cv;
      }
    }
    __syncthreads();
    #pragma unroll
    for (int off = 1; off < 32; off <<= 1) csum += __shfl_xor(csum, off, 32);
    if (lane == 0) commitbuf[qi * ROW_TILES + tile] = csum;  // deterministic slot
  }

  { // quantized = h - residual_final, overwrite LDS tile
    float4* R4 = (float4*)R;
    const float4* H4 = (const float4*)(h + (size_t)row0 * D_DIM);
    for (int i = lane; i < (16 * D_DIM) / 4; i += 32) {
      const float4 hv = H4[i]; const float4 rv = R4[i];
      float4 qv; qv.x = hv.x - rv.x; qv.y = hv.y - rv.y;
      qv.z = hv.z - rv.z; qv.w = hv.w - rv.w;
      R4[i] = qv;
    }
  }
  __syncthreads();

  // GEMM2: out[m][n] = sum_d quant[m][d] * Wout[n][d] + bout[n]
  for (int ng = 0; ng < H_DIM / 64; ++ng) {       // 16 groups of 4 N-tiles
    const int n0 = ng << 6;
    const float* bp[4];
    #pragma unroll
    for (int j = 0; j < 4; ++j) {
      bp[j] = Wout + (size_t)(n0 + (j << 4) + rlane) * D_DIM + khalf;
      __builtin_prefetch(bp[j], 0, 0);
    }
    v8f acc[4] = {{0,0,0,0,0,0,0,0},{0,0,0,0,0,0,0,0},
                  {0,0,0,0,0,0,0,0},{0,0,0,0,0,0,0,0}};
    #pragma unroll 4
    for (int k0 = 0; k0 < D_DIM; k0 += 4) {
      v2f a = ld2(aptr + k0);
      #pragma unroll
      for (int j = 0; j < 4; ++j)
        acc[j] = wmma_f32(a, ld2(bp[j] + k0), acc[j]);
    }
    const int rbase = row0 + ((lane >> 4) << 3);
    #pragma unroll
    for (int j = 0; j < 4; ++j) {
      const int col  = n0 + (j << 4) + rlane;
      const float bv = bout[col];
      #pragma unroll
      for (int p = 0; p < 8; ++p)
        out[(size_t)(rbase + p) * H_DIM + col] = acc[j][p] + bv;
    }
  }
}

// ---------------------------------------------------------------------------
// com = (sum over all commit partials) / (Q * M * D)   [COMMIT_W = 1.0]
// ---------------------------------------------------------------------------
__global__ __launch_bounds__(256) void commit_reduce_kernel(
    const float* __restrict__ commitbuf, float* __restrict__ comptr) {
  __shared__ float sh[256];
  float s = 0.f;
  for (int i = threadIdx.x; i < Q_NUM * ROW_TILES; i += 256) s += commitbuf[i];
  sh[threadIdx.x] = s;
  __syncthreads();
  for (int stride = 128; stride > 0; stride >>= 1) {
    if ((int)threadIdx.x < stride) sh[threadIdx.x] += sh[threadIdx.x + stride];
    __syncthreads();
  }
  if (threadIdx.x == 0)
    comptr[0] = sh[0] / ((float)Q_NUM * (float)M_ROWS * (float)D_DIM);
}

// ---------------------------------------------------------------------------
extern "C" void kernel_launch(void* const* d_in, const int* in_sizes, int n_in,
                              void* d_out, int out_size, void* d_ws, size_t ws_size,
                              hipStream_t stream) {
  const float* x    = (const float*)d_in[0];   // [8,4096,1024]
  const float* Win  = (const float*)d_in[1];   // [256,1024]
  const float* bin  = (const float*)d_in[2];   // [256]
  const float* Wout = (const float*)d_in[3];   // [1024,256]
  const float* bout = (const float*)d_in[4];   // [1024]
  const float* cb   = (const float*)d_in[5];   // [4,1024,256]

  float* out = (float*)d_out;                            // [8,4096,1024]
  float* q1  = out + (size_t)M_ROWS * H_DIM;             // [8,4096,256]
  float* q2  = q1 + (size_t)M_ROWS * D_DIM;              // [8,4096,256]
  float* com = q2 + (size_t)M_ROWS * D_DIM;              // scalar

  float* h         = (float*)d_ws;                       // 8,388,608 f32 (32 MB)
  float* cnorms    = h + (size_t)M_ROWS * D_DIM;         // 4096 f32
  float* commitbuf = cnorms + Q_NUM * K_CODES;           // 8192 f32

  cnorm_kernel<<<dim3((Q_NUM * K_CODES) / 256), dim3(256), 0, stream>>>(cb, cnorms);
  gemm1_kernel<<<dim3(ROW_TILES / 4), dim3(128), 0, stream>>>(x, Win, bin, h);
  vq_fused_kernel<<<dim3(ROW_TILES / 2), dim3(64), 0, stream>>>(
      h, cb, cnorms, Wout, bout, out, q1, q2, commitbuf);
  commit_reduce_kernel<<<dim3(1), dim3(256), 0, stream>>>(commitbuf, com);
}